// GraphConvLayer_42691974922546
// MI455X (gfx1250) — compile-verified
//
#include <hip/hip_runtime.h>
#include <stdint.h>

// Problem constants (match reference: B=256, N=64, C=128)
#define BB   256
#define NN   64
#define CC   128
#define NODES (BB * NN)            // 16384
#define EE   (BB * NN * NN)        // 1048576 edges
#define GCHUNK 8                   // rows per TDM gather op (32-bit index mode max)
#define WPB  4                     // waves per block in aggregate kernel (128 thr)
#define OFF_STRIDE 16512           // padded int stride for ws arrays

typedef __attribute__((ext_vector_type(4))) unsigned int v4u_;
typedef __attribute__((ext_vector_type(8))) int          v8i_;
typedef __attribute__((ext_vector_type(4))) int          v4i_;

// ---------------------------------------------------------------------------
// TDM gather: load `nrows` rows (128 f32 each) of e_params, selected by 32-bit
// row indices, packed contiguously into LDS at lds_addr. ISA §8 (D# descriptor,
// gather mode). Uses TENSORcnt; EXEC is ignored (one issue per wave).
// This toolchain's builtin is the 6-arg form:
//   (uint32x4 g0, int32x8 g1, int32x4 g2, int32x4 g3, int32x8 extra, i32 cpol)
// ---------------------------------------------------------------------------
__device__ __forceinline__ void tdm_gather_rows(uint32_t lds_addr,
                                                const float* ep_base,
                                                const int* ids, int nrows) {
  uint64_t ga = (uint64_t)(uintptr_t)ep_base;
  v4u_ g0;
  // count=1 | gather_index_size=32b (bit30) | gather_mode=1 (bit31)
  g0[0] = 1u | (1u << 30) | (1u << 31);
  g0[1] = (uint32_t)__builtin_amdgcn_readfirstlane((int)lds_addr);
  g0[2] = (uint32_t)__builtin_amdgcn_readfirstlane((int)(uint32_t)ga);
  g0[3] = (uint32_t)__builtin_amdgcn_readfirstlane(
              (int)(((uint32_t)(ga >> 32) & 0x01FFFFFFu) | (2u << 30))); // type=2
  v8i_ g1;
  g1[0] = (int)(2u << 16);                                   // data_size = 4B
  g1[1] = (int)((uint32_t)CC << 16);                         // tensor_dim0 = 128
  g1[2] = (int)(((uint32_t)EE & 0xFFFFu) << 16);             // tensor_dim1 lo16
  g1[3] = (int)((((uint32_t)EE >> 16) & 0xFFFFu)             // tensor_dim1 hi16
                | ((uint32_t)CC << 16));                     // tile_dim0 = 128
  g1[4] = __builtin_amdgcn_readfirstlane(nrows & 0xFFFF);    // tile_dim1 = #idx
  g1[5] = CC;                                                // dim0 stride = 128
  g1[6] = 0;
  g1[7] = 0;
  v4i_ g2 = { __builtin_amdgcn_readfirstlane(ids[0]),
              __builtin_amdgcn_readfirstlane(ids[1]),
              __builtin_amdgcn_readfirstlane(ids[2]),
              __builtin_amdgcn_readfirstlane(ids[3]) };
  v4i_ g3 = { __builtin_amdgcn_readfirstlane(ids[4]),
              __builtin_amdgcn_readfirstlane(ids[5]),
              __builtin_amdgcn_readfirstlane(ids[6]),
              __builtin_amdgcn_readfirstlane(ids[7]) };
  v8i_ gx = { 0, 0, 0, 0, 0, 0, 0, 0 };
  __builtin_amdgcn_tensor_load_to_lds(g0, g1, g2, g3, gx, 0);
}

// ---------------------------------------------------------------------------
// Pass 0: zero the count/offset array.
// ---------------------------------------------------------------------------
__global__ void k_zero(int* __restrict__ p, int n) {
  int i = blockIdx.x * blockDim.x + threadIdx.x;
  if (i < n) p[i] = 0;
}

// Pass 1: histogram of destination ids into off[1+i] (int atomics, L2-resident).
__global__ void k_hist(const int* __restrict__ ii, int* __restrict__ off) {
  int e = blockIdx.x * blockDim.x + threadIdx.x;   // exact grid: EE threads
  atomicAdd(&off[1 + ii[e]], 1);
}

// Pass 2: single-block scan -> CSR row pointers; cur[i] = start offset copy.
__global__ void k_scan(int* __restrict__ off, int* __restrict__ cur) {
  __shared__ int part[256];
  __shared__ int partx[256];
  const int t = threadIdx.x;
  const int base = t * (NODES / 256);              // 64 bins per thread
  int s = 0;
#pragma unroll 4
  for (int k = 0; k < NODES / 256; ++k) s += off[1 + base + k];
  part[t] = s;
  __syncthreads();
  if (t == 0) {
    int r = 0;
    for (int i = 0; i < 256; ++i) { partx[i] = r; r += part[i]; }
  }
  __syncthreads();
  int run = partx[t];
  for (int k = 0; k < NODES / 256; ++k) {
    int v = off[1 + base + k];
    cur[base + k] = run;          // exclusive start for node (fill cursor)
    run += v;
    off[1 + base + k] = run;      // inclusive -> off[] becomes row pointers
  }
}

// Pass 3: bucket fill (counting sort of edge ids by destination).
__global__ void k_fill(const int* __restrict__ ii, int* __restrict__ cur,
                       int* __restrict__ bucket) {
  int e = blockIdx.x * blockDim.x + threadIdx.x;   // exact grid: EE threads
  int pos = atomicAdd(&cur[ii[e]], 1);
  bucket[pos] = e;
}

// ---------------------------------------------------------------------------
// Pass 4: pull aggregation. One wave32 per destination node; lane owns 4
// channels (float4 accumulator). e_params rows are staged into per-wave
// double-buffered LDS via TDM gather ops, pipelined with s_wait_tensorcnt.
// ---------------------------------------------------------------------------
__global__ void __launch_bounds__(WPB * 32)
k_aggregate(const float* __restrict__ flat, const float* __restrict__ ew,
            const float* __restrict__ ep,   const int* __restrict__ jj,
            const int* __restrict__ bucket, const int* __restrict__ off,
            float* __restrict__ out) {
  __shared__ float smem[WPB * 2 * GCHUNK * CC];    // 4 waves * 2 bufs * 4KB = 32KB
  const int lane = threadIdx.x & 31;
  const int wave = threadIdx.x >> 5;
  const int node = blockIdx.x * WPB + wave;

  const int beg = off[node];
  const int end = off[node + 1];

  const uint32_t lds_w =
      (uint32_t)(uintptr_t)(void*)&smem[wave * 2 * GCHUNK * CC];

  float4 acc; acc.x = 0.f; acc.y = 0.f; acc.z = 0.f; acc.w = 0.f;

  int ids[GCHUNK];
  int pos = beg;
  int cnt = end - pos; if (cnt > GCHUNK) cnt = GCHUNK;
  if (cnt > 0) {
#pragma unroll
    for (int k = 0; k < GCHUNK; ++k) ids[k] = (k < cnt) ? bucket[pos + k] : 0;
    tdm_gather_rows(lds_w, ep, ids, cnt);          // prologue: fill buf 0
  }

  int buf = 0;
  while (cnt > 0) {
    const int npos = pos + cnt;
    int ncnt = end - npos; if (ncnt > GCHUNK) ncnt = GCHUNK; if (ncnt < 0) ncnt = 0;
    int nids[GCHUNK];
    if (ncnt > 0) {
#pragma unroll
      for (int k = 0; k < GCHUNK; ++k) nids[k] = (k < ncnt) ? bucket[npos + k] : 0;
      tdm_gather_rows(lds_w + (uint32_t)((buf ^ 1) * GCHUNK * CC * 4), ep,
                      nids, ncnt);                 // prefetch next chunk
      __builtin_amdgcn_s_wait_tensorcnt(1);        // current chunk resident
    } else {
#pragma unroll
      for (int k = 0; k < GCHUNK; ++k) nids[k] = 0;
      __builtin_amdgcn_s_wait_tensorcnt(0);        // drain last chunk
    }

    const float* sp = &smem[(wave * 2 + buf) * GCHUNK * CC];
#pragma unroll
    for (int k = 0; k < GCHUNK; ++k) {
      if (k < cnt) {
        const int   e = ids[k];
        const float w = ew[e];                     // per-edge softmax scalar
        const int   j = jj[e];                     // source node
        const float4 p = *(const float4*)(sp + k * CC + (lane << 2));       // ds b128
        const float4 f = *(const float4*)(flat + (size_t)j * CC + (lane << 2)); // L2 hit
        acc.x = fmaf(w * p.x, f.x, acc.x);
        acc.y = fmaf(w * p.y, f.y, acc.y);
        acc.z = fmaf(w * p.z, f.z, acc.z);
        acc.w = fmaf(w * p.w, f.w, acc.w);
      }
    }
#pragma unroll
    for (int k = 0; k < GCHUNK; ++k) ids[k] = nids[k];
    pos = npos; cnt = ncnt; buf ^= 1;
  }

  // Fused ReLU + store; untouched nodes get exact zeros (matches segment_sum).
  float4 r;
  r.x = fmaxf(acc.x, 0.f); r.y = fmaxf(acc.y, 0.f);
  r.z = fmaxf(acc.z, 0.f); r.w = fmaxf(acc.w, 0.f);
  *(float4*)(out + (size_t)node * CC + (lane << 2)) = r;
}

// ---------------------------------------------------------------------------
extern "C" void kernel_launch(void* const* d_in, const int* in_sizes, int n_in,
                              void* d_out, int out_size, void* d_ws, size_t ws_size,
                              hipStream_t stream) {
  (void)in_sizes; (void)n_in; (void)out_size; (void)ws_size;
  const float* n_feats = (const float*)d_in[0];   // (B*N, C) after reshape
  const float* e_w     = (const float*)d_in[1];   // (E,)
  const float* e_p     = (const float*)d_in[2];   // (E, C)
  const int*   ii      = (const int*)d_in[3];     // dest node ids
  const int*   jj      = (const int*)d_in[4];     // src node ids
  float* out = (float*)d_out;

  // Workspace layout (ints): off[16385] | cur[16384] | bucket[E]  (~4.3 MB)
  int* off    = (int*)d_ws;
  int* cur    = off + OFF_STRIDE;
  int* bucket = cur + OFF_STRIDE;

  k_zero<<<(NODES + 1 + 255) / 256, 256, 0, stream>>>(off, NODES + 1);
  k_hist<<<EE / 256, 256, 0, stream>>>(ii, off);
  k_scan<<<1, 256, 0, stream>>>(off, cur);
  k_fill<<<EE / 256, 256, 0, stream>>>(ii, cur, bucket);
  k_aggregate<<<NODES / WPB, WPB * 32, 0, stream>>>(n_feats, e_w, e_p, jj,
                                                    bucket, off, out);
}